// TinyGRUInt8_22780506538548
// MI455X (gfx1250) — compile-verified
//
#include <hip/hip_runtime.h>

// ---------------------------------------------------------------------------
// Fused GRU (B=2048, T=1024, C=9, H=36) + linear head for MI455X (gfx1250).
//
// One wave32 per 16-batch-row tile. h kept resident across all T=1024 steps.
// f32 WMMA 16x16x4: M=16 batch rows, K=36 hidden (9 chunks of 4), N = gate
// columns padded 3*36 -> 3*48 = 144 (9 tiles) so r/z/n align per lane.
// Input GEMM (K=9 padded to 12, 3 chunks) fused into the same accumulators
// for r/z; kept separate for the n gate (r * (W_hn h + b_hn) semantics).
// ---------------------------------------------------------------------------

typedef __attribute__((ext_vector_type(2))) float v2f;
typedef __attribute__((ext_vector_type(8))) float v8f;

#define B_SZ   2048
#define T_SZ   1024
#define C_SZ   9
#define H_SZ   36
#define PG     48          // padded per-gate width
#define NT     9           // N tiles (144 / 16)
#define KC_H   9           // K chunks for hidden GEMM (36/4)
#define KC_X   3           // K chunks for input GEMM (12/4, K padded 9->12)
#define HSTR   44          // LDS row stride for h transpose buffer (floats)

__device__ __forceinline__ float fast_sigmoid(float a) {
    return 1.0f / (1.0f + __expf(-a));
}
__device__ __forceinline__ float fast_tanh(float a) {
    // tanh(a) = 1 - 2/(exp(2a)+1)
    return 1.0f - 2.0f / (__expf(2.0f * a) + 1.0f);
}

__launch_bounds__(32)
__global__ void gru_fused_wmma(const float* __restrict__ x,
                               const float* __restrict__ w_ih,
                               const float* __restrict__ w_hh,
                               const float* __restrict__ b_ih,
                               const float* __restrict__ b_hh,
                               const float* __restrict__ w_head,
                               const float* __restrict__ b_head,
                               float* __restrict__ out)
{
    // Pre-swizzled B operands: sB*[tile*32 + lane] = (reg0, reg1) for that lane.
    // B-matrix 4x16 f32 layout (mirrors documented A 16x4 layout):
    //   reg j, lanes 0-15  -> K = kc*4 + j,     N = lane
    //   reg j, lanes 16-31 -> K = kc*4 + j + 2, N = lane-16
    __shared__ v2f  sBhh[KC_H * NT * 32];   // 20.25 KB
    __shared__ v2f  sBih[KC_X * NT * 32];   //  6.75 KB
    __shared__ float sH[16 * HSTR];         //  2.75 KB  h transpose buffer

    const int lane = threadIdx.x;         // 0..31
    const int half = lane >> 4;           // 0 | 1
    const int ln   = lane & 15;
    const int b0   = blockIdx.x * 16;

    // ---- stage swizzled hidden-weight B operands (padded gate columns) ----
    for (int tile = lane == lane ? 0 : 0; tile < KC_H * NT; ++tile) {
        const int kc = tile / NT, nt = tile % NT;
        const int p  = nt * 16 + ln;           // padded gate column 0..143
        const int g  = p / PG, jj = p % PG;    // gate id, col within gate
        const int k0 = kc * 4 + half * 2;      // hidden index (K)
        v2f v; v.x = 0.0f; v.y = 0.0f;
        if (jj < H_SZ) {
            const int col = g * H_SZ + jj;     // original gate column 0..107
            v.x = w_hh[col * H_SZ + k0];
            v.y = w_hh[col * H_SZ + k0 + 1];
        }
        sBhh[tile * 32 + lane] = v;
    }
    // ---- stage swizzled input-weight B operands (K padded 9 -> 12) ----
    for (int tile = 0; tile < KC_X * NT; ++tile) {
        const int kc = tile / NT, nt = tile % NT;
        const int p  = nt * 16 + ln;
        const int g  = p / PG, jj = p % PG;
        const int k0 = kc * 4 + half * 2;
        v2f v; v.x = 0.0f; v.y = 0.0f;
        if (jj < H_SZ) {
            const int col = g * H_SZ + jj;
            if (k0     < C_SZ) v.x = w_ih[col * C_SZ + k0];
            if (k0 + 1 < C_SZ) v.y = w_ih[col * C_SZ + k0 + 1];
        }
        sBih[tile * 32 + lane] = v;
    }
    // ---- per-lane biases in padded-column space ----
    float bx[NT], bh[NT];
#pragma unroll
    for (int nt = 0; nt < NT; ++nt) {
        const int p = nt * 16 + ln;
        const int g = p / PG, jj = p % PG;
        float vx = 0.0f, vh = 0.0f;
        if (jj < H_SZ) { vx = b_ih[g * H_SZ + jj]; vh = b_hh[g * H_SZ + jj]; }
        bx[nt] = vx; bh[nt] = vh;
    }
    // zero h transpose buffer (h0 = 0)
    for (int i = lane; i < 16 * HSTR; i += 32) sH[i] = 0.0f;
    __syncthreads();   // single wave -> NOP, but orders LDS for the compiler

    const float* xrow = x + (size_t)(b0 + ln) * T_SZ * C_SZ;

    // h in C/D layout: hC[k][r] = h[row = r + half*8][col = k*16 + ln]
    v8f hC[3];
#pragma unroll
    for (int k = 0; k < 3; ++k)
        hC[k] = (v8f){0.f,0.f,0.f,0.f,0.f,0.f,0.f,0.f};

    for (int t = 0; t < T_SZ; ++t) {
        // -- h in A layout (16x4 chunks): lane holds row ln, cols kc*4+half*2.. --
        v2f hA[KC_H];
#pragma unroll
        for (int kc = 0; kc < KC_H; ++kc) {
            const int c0 = kc * 4 + half * 2;
            hA[kc].x = sH[ln * HSTR + c0];
            hA[kc].y = sH[ln * HSTR + c0 + 1];
        }
        // -- x_t in A layout (16x12, zero padded cols 9..11) --
        v2f xA[KC_X];
#pragma unroll
        for (int kc = 0; kc < KC_X; ++kc) {
            const int c0 = kc * 4 + half * 2;
            float v0 = 0.0f, v1 = 0.0f;
            if (c0     < C_SZ) v0 = xrow[t * C_SZ + c0];
            if (c0 + 1 < C_SZ) v1 = xrow[t * C_SZ + c0 + 1];
            xA[kc].x = v0; xA[kc].y = v1;
        }

        // -- r,z tiles (0..5): gx and gh accumulate into the same C --
        v8f acc[6];
#pragma unroll
        for (int nt = 0; nt < 6; ++nt) {
            v8f c = (v8f){0.f,0.f,0.f,0.f,0.f,0.f,0.f,0.f};
#pragma unroll
            for (int kc = 0; kc < KC_X; ++kc)
                c = __builtin_amdgcn_wmma_f32_16x16x4_f32(
                        false, xA[kc], false, sBih[(kc * NT + nt) * 32 + lane],
                        (short)0, c, false, false);
#pragma unroll
            for (int kc = 0; kc < KC_H; ++kc)
                c = __builtin_amdgcn_wmma_f32_16x16x4_f32(
                        false, hA[kc], false, sBhh[(kc * NT + nt) * 32 + lane],
                        (short)0, c, false, false);
            acc[nt] = c;
        }
        // -- n tiles (6..8): keep input and hidden contributions separate --
        v8f gxn[3], ghn[3];
#pragma unroll
        for (int k = 0; k < 3; ++k) {
            const int nt = 6 + k;
            v8f c = (v8f){0.f,0.f,0.f,0.f,0.f,0.f,0.f,0.f};
#pragma unroll
            for (int kc = 0; kc < KC_X; ++kc)
                c = __builtin_amdgcn_wmma_f32_16x16x4_f32(
                        false, xA[kc], false, sBih[(kc * NT + nt) * 32 + lane],
                        (short)0, c, false, false);
            gxn[k] = c;
            v8f d = (v8f){0.f,0.f,0.f,0.f,0.f,0.f,0.f,0.f};
#pragma unroll
            for (int kc = 0; kc < KC_H; ++kc)
                d = __builtin_amdgcn_wmma_f32_16x16x4_f32(
                        false, hA[kc], false, sBhh[(kc * NT + nt) * 32 + lane],
                        (short)0, d, false, false);
            ghn[k] = d;
        }

        // -- gate nonlinearity + h update; r/z/n are at tiles k, k+3, k+6,
        //    same lane, same register => pure per-lane VALU --
#pragma unroll
        for (int k = 0; k < 3; ++k) {
#pragma unroll
            for (int r8 = 0; r8 < 8; ++r8) {
                const float gr = acc[k][r8]     + bx[k]     + bh[k];
                const float gz = acc[3 + k][r8] + bx[3 + k] + bh[3 + k];
                const float rg = fast_sigmoid(gr);
                const float zg = fast_sigmoid(gz);
                const float nn = fast_tanh(gxn[k][r8] + bx[6 + k] +
                                           rg * (ghn[k][r8] + bh[6 + k]));
                hC[k][r8] = (1.0f - zg) * nn + zg * hC[k][r8];
            }
        }

        // -- transpose h back to [row][col] in LDS for next step's A operand --
#pragma unroll
        for (int k = 0; k < 3; ++k) {
            const int col = k * 16 + ln;
            if (col < H_SZ) {
#pragma unroll
                for (int r8 = 0; r8 < 8; ++r8)
                    sH[(r8 + half * 8) * HSTR + col] = hC[k][r8];
            }
        }
        __syncthreads();
    }

    // ---- linear head: out[b] = h . w_head + b_head ----
    if (lane < 16) {
        float a = b_head[0];
#pragma unroll
        for (int j = 0; j < H_SZ; ++j)
            a += sH[ln * HSTR + j] * w_head[j];
        out[b0 + ln] = a;
    }
}

extern "C" void kernel_launch(void* const* d_in, const int* in_sizes, int n_in,
                              void* d_out, int out_size, void* d_ws, size_t ws_size,
                              hipStream_t stream) {
    (void)in_sizes; (void)n_in; (void)d_ws; (void)ws_size; (void)out_size;
    const float* x      = (const float*)d_in[0];
    const float* w_ih   = (const float*)d_in[1];
    const float* w_hh   = (const float*)d_in[2];
    const float* b_ih   = (const float*)d_in[3];
    const float* b_hh   = (const float*)d_in[4];
    const float* w_head = (const float*)d_in[5];
    const float* b_head = (const float*)d_in[6];
    float* out = (float*)d_out;

    gru_fused_wmma<<<dim3(B_SZ / 16), dim3(32), 0, stream>>>(
        x, w_ih, w_hh, b_ih, b_hh, w_head, b_head, out);
}